// OperationEmbeddingLayer_74217034875541
// MI455X (gfx1250) — compile-verified
//
#include <hip/hip_runtime.h>

// fp32 WMMA fragment types for V_WMMA_F32_16X16X4_F32 (wave32):
//   A: 16x4 f32  -> 2 VGPRs (v2f)   B: 4x16 f32 -> 2 VGPRs (v2f)   C/D: 16x16 f32 -> 8 VGPRs (v8f)
typedef float v2f __attribute__((ext_vector_type(2)));
typedef float v8f __attribute__((ext_vector_type(8)));

// ---------------------------------------------------------------------------
// Edge scatter-sum:  out[dstIdx[e], :] += src[srcIdx[e], :]
// One thread per (edge, 4-float chunk); float4 source read, 4x f32 atomics.
// Agg buffers (70 MB) are L2-resident on MI455X (192 MB L2), so the atomics
// stay in-cache; HBM cost is the 512B-row gathers, same as any CSR scheme.
// ---------------------------------------------------------------------------
template<int D>
__global__ __launch_bounds__(256) void scatter_add_kernel(
    const float* __restrict__ src, const int* __restrict__ srcIdx,
    const int* __restrict__ dstIdx, float* __restrict__ out, int E)
{
    constexpr int TPE = D / 4;
    long t = (long)blockIdx.x * blockDim.x + threadIdx.x;
    int e = (int)(t / TPE);
    if (e >= E) return;
    int f = (int)(t % TPE) * 4;
    int s = srcIdx[e];
    int d = dstIdx[e];
    const float4 v = *(const float4*)(src + (long)s * D + f);
    float* o = out + (long)d * D + f;
    atomicAdd(o + 0, v.x);
    atomicAdd(o + 1, v.y);
    atomicAdd(o + 2, v.z);
    atomicAdd(o + 3, v.w);
}

// ---------------------------------------------------------------------------
// Fused GEMM + bias (+ ReLU):  Out[r, c] = act( A[gather(r), :] @ W + b )[c]
//   A: [*, K] contiguous, W: [K, M] row-major, Out row stride ldo.
// Block: 256 threads = 8 wave32, covers ROWS=80 rows x M cols.
// K consumed in 32-wide chunks staged in padded LDS; accumulators live in
// VGPRs across chunks. Each B fragment is reused for 5 row-tile WMMAs,
// cutting W (L2) traffic 5x vs a 16-row block and raising WMMA/vmem density.
// ---------------------------------------------------------------------------
template<int K, int M, bool RELU>
__global__ __launch_bounds__(256) void gemm_wmma_kernel(
    const float* __restrict__ A,
    const int*   __restrict__ gather,   // nullptr => identity
    const float* __restrict__ W,
    const float* __restrict__ bias,
    float* __restrict__ Out,
    int ldo)
{
    constexpr int ROWS   = 80;          // rows per block (N = 625 * 80 exactly)
    constexpr int RT     = ROWS / 16;   // 5 row tiles
    constexpr int KC     = 32;          // K chunk (all K are multiples of 32)
    constexpr int KCP    = KC + 4;      // padded LDS row stride (floats)
    constexpr int CTILES = M / 16;
    constexpr int CTMAX  = (CTILES + 7) / 8;

    __shared__ float As[ROWS * KCP];
    __shared__ int   gidx[ROWS];

    const int tid     = threadIdx.x;
    const int rowbase = blockIdx.x * ROWS;

    if (tid < ROWS)
        gidx[tid] = gather ? gather[rowbase + tid] : (rowbase + tid);
    __syncthreads();

    const int wave = tid >> 5;
    const int lane = tid & 31;
    const int half = lane >> 4;         // 0: lanes 0-15, 1: lanes 16-31
    const int l    = lane & 15;
    const int koff = half * 2;          // K sub-offset carried by this half-wave

    v8f acc[CTMAX][RT];
    #pragma unroll
    for (int ci = 0; ci < CTMAX; ++ci)
        #pragma unroll
        for (int rt = 0; rt < RT; ++rt)
            acc[ci][rt] = (v8f){0.f,0.f,0.f,0.f,0.f,0.f,0.f,0.f};

    for (int kc = 0; kc < K; kc += KC) {
        // Stage ROWS x KC of A into LDS (float4 granularity, coalesced).
        constexpr int VECS = ROWS * KC / 4;
        for (int i = tid; i < VECS; i += 256) {
            int r  = i / (KC / 4);
            int c4 = (i % (KC / 4)) * 4;
            const float4 v = *(const float4*)(A + (long)gidx[r] * K + kc + c4);
            *(float4*)(As + r * KCP + c4) = v;
        }
        __syncthreads();

        #pragma unroll
        for (int ks = 0; ks < KC; ks += 4) {
            // A fragments for all 5 row tiles (shared across column tiles).
            // A 16x4: lanes 0-15 hold (K=k+0,k+1); lanes 16-31 hold (k+2,k+3)
            v2f a[RT];
            #pragma unroll
            for (int rt = 0; rt < RT; ++rt) {
                const float* Arow = As + (rt * 16 + l) * KCP + ks + koff;
                a[rt].x = Arow[0];
                a[rt].y = Arow[1];
            }
            #pragma unroll
            for (int ci = 0; ci < CTMAX; ++ci) {
                const int ct = wave + ci * 8;
                if (ct < CTILES) {              // wave-uniform: EXEC stays full
                    const int cb = ct * 16 + l;
                    // B 4x16: VGPR0 rows k+0/k+2 across half-waves, VGPR1 k+1/k+3
                    v2f b;
                    b.x = W[(long)(kc + ks + koff)     * M + cb];
                    b.y = W[(long)(kc + ks + koff + 1) * M + cb];
                    #pragma unroll
                    for (int rt = 0; rt < RT; ++rt)
                        acc[ci][rt] = __builtin_amdgcn_wmma_f32_16x16x4_f32(
                            false, a[rt], false, b, (short)0, acc[ci][rt],
                            false, false);
                }
            }
        }
        __syncthreads();
    }

    // Epilogue: bias (+ReLU) and strided store.
    // C/D layout: VGPR g -> row base+g (lanes 0-15) / base+8+g (lanes 16-31)
    #pragma unroll
    for (int ci = 0; ci < CTMAX; ++ci) {
        const int ct = wave + ci * 8;
        if (ct < CTILES) {
            const float bv = bias[ct * 16 + l];
            #pragma unroll
            for (int rt = 0; rt < RT; ++rt) {
                #pragma unroll
                for (int g = 0; g < 8; ++g) {
                    float v = acc[ci][rt][g] + bv;
                    if (RELU) v = v > 0.f ? v : 0.f;
                    const long row = rowbase + rt * 16 + g + half * 8;
                    Out[row * ldo + ct * 16 + l] = v;
                }
            }
        }
    }
}

// ---------------------------------------------------------------------------
extern "C" void kernel_launch(void* const* d_in, const int* in_sizes, int n_in,
                              void* d_out, int out_size, void* d_ws, size_t ws_size,
                              hipStream_t stream) {
    (void)n_in; (void)out_size; (void)ws_size;

    const float* operations = (const float*)d_in[0];   // [N,128]
    const float* items      = (const float*)d_in[1];   // [NI,128]
    const float* materials  = (const float*)d_in[2];   // [NM,32]
    const float* resources  = (const float*)d_in[3];   // [NR,64]
    const int*   related    = (const int*)d_in[4];     // [N]
    const int*   res_op     = (const int*)d_in[5];     // [E_RES]
    const int*   res_res    = (const int*)d_in[6];
    const int*   mat_op     = (const int*)d_in[7];     // [E_MAT]
    const int*   mat_mat    = (const int*)d_in[8];
    const int*   prec_src   = (const int*)d_in[9];     // [E_PREC]
    const int*   prec_dst   = (const int*)d_in[10];

    const float* w_self1 = (const float*)d_in[11]; const float* b_self1 = (const float*)d_in[12];
    const float* w_self2 = (const float*)d_in[13]; const float* b_self2 = (const float*)d_in[14];
    const float* w_item1 = (const float*)d_in[15]; const float* b_item1 = (const float*)d_in[16];
    const float* w_item2 = (const float*)d_in[17]; const float* b_item2 = (const float*)d_in[18];
    const float* w_pred1 = (const float*)d_in[19]; const float* b_pred1 = (const float*)d_in[20];
    const float* w_pred2 = (const float*)d_in[21]; const float* b_pred2 = (const float*)d_in[22];
    const float* w_succ1 = (const float*)d_in[23]; const float* b_succ1 = (const float*)d_in[24];
    const float* w_succ2 = (const float*)d_in[25]; const float* b_succ2 = (const float*)d_in[26];
    const float* w_res1  = (const float*)d_in[27]; const float* b_res1  = (const float*)d_in[28];
    const float* w_res2  = (const float*)d_in[29]; const float* b_res2  = (const float*)d_in[30];
    const float* w_mat1  = (const float*)d_in[31]; const float* b_mat1  = (const float*)d_in[32];
    const float* w_mat2  = (const float*)d_in[33]; const float* b_mat2  = (const float*)d_in[34];
    const float* w_c1    = (const float*)d_in[35]; const float* b_c1    = (const float*)d_in[36];
    const float* w_c2    = (const float*)d_in[37]; const float* b_c2    = (const float*)d_in[38];
    const float* w_c3    = (const float*)d_in[39]; const float* b_c3    = (const float*)d_in[40];

    const long N      = in_sizes[4];   // 50000
    const int  E_RES  = in_sizes[5];   // 100000
    const int  E_MAT  = in_sizes[7];   // 100000
    const int  E_PREC = in_sizes[9];   // 800000

    // Workspace layout (floats):
    //   X   [N,608] concat: pred(0) succ(128) res(256) mat(320) item(352) self(480)
    //   H1  [N,256] shared hidden buffer (stream-ordered reuse)
    //   AGG [N,352] = aggPred(128) aggSucc(128) aggRes(64) aggMat(32); aliased as H2 later
    float* ws      = (float*)d_ws;
    float* X       = ws;
    float* H1      = X + N * 608;
    float* AGG     = H1 + N * 256;
    float* aggPred = AGG;
    float* aggSucc = aggPred + N * 128;
    float* aggRes  = aggSucc + N * 128;
    float* aggMat  = aggRes  + N * 64;
    float* H2      = AGG;              // [N,128], reused after embeddings are in X

    // 1) zero aggregation buffers (capture-safe async memset)
    hipMemsetAsync(AGG, 0, (size_t)N * 352 * sizeof(float), stream);

    // 2) edge scatter-sums
    {
        long T = (long)E_MAT * (32 / 4);
        scatter_add_kernel<32><<<dim3((T + 255) / 256), dim3(256), 0, stream>>>(
            materials, mat_mat, mat_op, aggMat, E_MAT);
    }
    {
        long T = (long)E_RES * (64 / 4);
        scatter_add_kernel<64><<<dim3((T + 255) / 256), dim3(256), 0, stream>>>(
            resources, res_res, res_op, aggRes, E_RES);
    }
    {
        long T = (long)E_PREC * (128 / 4);
        // agg_pred = segment_sum(operations[prec_dst] -> prec_src)
        scatter_add_kernel<128><<<dim3((T + 255) / 256), dim3(256), 0, stream>>>(
            operations, prec_dst, prec_src, aggPred, E_PREC);
        // agg_succ = segment_sum(operations[prec_src] -> prec_dst)
        scatter_add_kernel<128><<<dim3((T + 255) / 256), dim3(256), 0, stream>>>(
            operations, prec_src, prec_dst, aggSucc, E_PREC);
    }

    const dim3 gGrid((unsigned)(N / 80)), gBlk(256);

    // 3) six MLP2s, each: [K->256 ReLU] then [256->M] written into its X slice
    // self
    gemm_wmma_kernel<128, 256, true ><<<gGrid, gBlk, 0, stream>>>(operations, nullptr, w_self1, b_self1, H1, 256);
    gemm_wmma_kernel<256, 128, false><<<gGrid, gBlk, 0, stream>>>(H1, nullptr, w_self2, b_self2, X + 480, 608);
    // item (gathered rows)
    gemm_wmma_kernel<128, 256, true ><<<gGrid, gBlk, 0, stream>>>(items, related, w_item1, b_item1, H1, 256);
    gemm_wmma_kernel<256, 128, false><<<gGrid, gBlk, 0, stream>>>(H1, nullptr, w_item2, b_item2, X + 352, 608);
    // mat
    gemm_wmma_kernel< 32, 256, true ><<<gGrid, gBlk, 0, stream>>>(aggMat, nullptr, w_mat1, b_mat1, H1, 256);
    gemm_wmma_kernel<256,  32, false><<<gGrid, gBlk, 0, stream>>>(H1, nullptr, w_mat2, b_mat2, X + 320, 608);
    // res
    gemm_wmma_kernel< 64, 256, true ><<<gGrid, gBlk, 0, stream>>>(aggRes, nullptr, w_res1, b_res1, H1, 256);
    gemm_wmma_kernel<256,  64, false><<<gGrid, gBlk, 0, stream>>>(H1, nullptr, w_res2, b_res2, X + 256, 608);
    // pred
    gemm_wmma_kernel<128, 256, true ><<<gGrid, gBlk, 0, stream>>>(aggPred, nullptr, w_pred1, b_pred1, H1, 256);
    gemm_wmma_kernel<256, 128, false><<<gGrid, gBlk, 0, stream>>>(H1, nullptr, w_pred2, b_pred2, X + 0, 608);
    // succ
    gemm_wmma_kernel<128, 256, true ><<<gGrid, gBlk, 0, stream>>>(aggSucc, nullptr, w_succ1, b_succ1, H1, 256);
    gemm_wmma_kernel<256, 128, false><<<gGrid, gBlk, 0, stream>>>(H1, nullptr, w_succ2, b_succ2, X + 128, 608);

    // 4) combiner head: 608 -> 256 ReLU -> 128 ReLU -> 128
    gemm_wmma_kernel<608, 256, true ><<<gGrid, gBlk, 0, stream>>>(X,  nullptr, w_c1, b_c1, H1, 256);
    gemm_wmma_kernel<256, 128, true ><<<gGrid, gBlk, 0, stream>>>(H1, nullptr, w_c2, b_c2, H2, 128);
    gemm_wmma_kernel<128, 128, false><<<gGrid, gBlk, 0, stream>>>(H2, nullptr, w_c3, b_c3, (float*)d_out, 128);
}